// RSoftmax_50620484551248
// MI455X (gfx1250) — compile-verified
//
#include <hip/hip_runtime.h>
#include <stdint.h>

#ifndef __has_builtin
#define __has_builtin(x) 0
#endif

#if __has_builtin(__builtin_amdgcn_global_load_async_to_lds_b128)
#define USE_ASYNC_LDS 1
#else
#define USE_ASYNC_LDS 0
#endif

#define FEAT 8192
#define TPB  512
#define WAVES (TPB / 32)
#define VPT  4              // float4 vectors per thread
#define EPT  (VPT * 4)      // 16 f32 elements per thread  (512*16 = 8192)

typedef int v4i __attribute__((ext_vector_type(4)));
typedef __attribute__((address_space(1))) v4i gv4i;   // global int4
typedef __attribute__((address_space(3))) v4i lv4i;   // LDS int4

__device__ __forceinline__ gv4i* as_global_v4(const void* p) {
  // generic->global: integer round-trip is a plain addrspace reinterpret on amdgcn
  return (gv4i*)(unsigned long long)(uintptr_t)p;
}
__device__ __forceinline__ lv4i* as_lds_v4(const void* p) {
  // generic LDS pointer = {shared aperture | 32-bit LDS offset}; truncation yields as(3) offset
  return (lv4i*)(unsigned)(uintptr_t)p;
}

// order-preserving f32 -> u32 key
__device__ __forceinline__ uint32_t f2key(float x) {
  uint32_t s = __float_as_uint(x);
  uint32_t m = (uint32_t)((int32_t)s >> 31) | 0x80000000u;
  return s ^ m;
}
__device__ __forceinline__ float key2f(uint32_t u) {
  uint32_t m = (u & 0x80000000u) ? 0x80000000u : 0xFFFFFFFFu;
  return __uint_as_float(u ^ m);
}

__device__ __forceinline__ int waveRedI(int v) {
#pragma unroll
  for (int o = 16; o > 0; o >>= 1) v += __shfl_xor(v, o, 32);
  return v;
}
__device__ __forceinline__ float waveRedF(float v) {
#pragma unroll
  for (int o = 16; o > 0; o >>= 1) v += __shfl_xor(v, o, 32);
  return v;
}

__global__ void __launch_bounds__(TPB)
rsoftmax_kernel(const float* __restrict__ in, const float* __restrict__ rate_p,
                float* __restrict__ out) {
  const int row  = blockIdx.x;
  const int tid  = threadIdx.x;
  const int lane = tid & 31;
  const int wave = tid >> 5;

  const float* rp = in  + (size_t)row * FEAT;
  float*       op = out + (size_t)row * FEAT;

#if USE_ASYNC_LDS
  __shared__ __align__(16) float s_row[FEAT];
#endif
  __shared__ int   s_parti[WAVES];
  __shared__ float s_partf[WAVES];

  // rank of the threshold element (matches (rate*n).astype(int32) + take-clamp)
  float rate = rate_p[0];
  rate = fminf(fmaxf(rate, 0.0f), 1.0f);
  int idx = (int)(rate * (float)FEAT);
  if (idx > FEAT - 1) idx = FEAT - 1;

  float x[EPT];

#if USE_ASYNC_LDS
  // Stage the whole row into LDS with the CDNA5 async copy engine (ASYNCcnt path).
#pragma unroll
  for (int j = 0; j < VPT; ++j) {
    const int v = tid + TPB * j;                       // float4 index in row
    __builtin_amdgcn_global_load_async_to_lds_b128(
        as_global_v4(rp + 4 * v), as_lds_v4(&s_row[4 * v]), 0, 0);
  }
#if __has_builtin(__builtin_amdgcn_s_wait_asynccnt)
  __builtin_amdgcn_s_wait_asynccnt(0);
#else
  asm volatile("s_wait_asynccnt 0x0" ::: "memory");
#endif
  __syncthreads();                                     // all waves' copies visible
#pragma unroll
  for (int j = 0; j < VPT; ++j) {
    const int v = tid + TPB * j;
    float4 t = *reinterpret_cast<const float4*>(&s_row[4 * v]);  // ds_load_b128
    x[4 * j + 0] = t.x; x[4 * j + 1] = t.y;
    x[4 * j + 2] = t.z; x[4 * j + 3] = t.w;
  }
#else
  // Fallback: direct coalesced global_load_b128 into registers.
#pragma unroll
  for (int j = 0; j < VPT; ++j) {
    const int v = tid + TPB * j;
    float4 t = *reinterpret_cast<const float4*>(rp + 4 * v);
    x[4 * j + 0] = t.x; x[4 * j + 1] = t.y;
    x[4 * j + 2] = t.z; x[4 * j + 3] = t.w;
  }
#endif

  // ---- exact k-th largest via 32-step bisection on the u32 key space ----
  uint32_t k[EPT];
#pragma unroll
  for (int e = 0; e < EPT; ++e) k[e] = f2key(x[e]);

  uint32_t lo = 0u, hi = 0xFFFFFFFFu;
  while (lo < hi) {                       // uniform across block (<= 32 iters)
    const uint32_t mid = lo + ((hi - lo) >> 1);
    int c = 0;
#pragma unroll
    for (int e = 0; e < EPT; ++e) c += (k[e] > mid) ? 1 : 0;
    c = waveRedI(c);
    if (lane == 0) s_parti[wave] = c;
    __syncthreads();
    int total = 0;
#pragma unroll
    for (int w = 0; w < WAVES; ++w) total += s_parti[w];   // broadcast reads
    __syncthreads();                      // protect s_parti before next write
    if (total > idx) lo = mid + 1; else hi = mid;
  }
  const float thresh = key2f(lo);         // == sorted_desc[idx]

  // ---- weighted exp + row sum ----
  float we[EPT];
  float psum = 0.0f;
#pragma unroll
  for (int e = 0; e < EPT; ++e) {
    float w = x[e] - thresh;
    w = w > 0.0f ? w : 0.0f;              // relu(x - thresh)
    const float v = w * expf(x[e]);
    we[e] = v;
    psum += v;
  }
  psum = waveRedF(psum);
  if (lane == 0) s_partf[wave] = psum;
  __syncthreads();
  float tot = 0.0f;
#pragma unroll
  for (int w = 0; w < WAVES; ++w) tot += s_partf[w];
  const float inv = 1.0f / tot;

  // ---- normalized output, global_store_b128 ----
#pragma unroll
  for (int j = 0; j < VPT; ++j) {
    const int v = tid + TPB * j;
    float4 o;
    o.x = we[4 * j + 0] * inv;
    o.y = we[4 * j + 1] * inv;
    o.z = we[4 * j + 2] * inv;
    o.w = we[4 * j + 3] * inv;
    *reinterpret_cast<float4*>(op + 4 * v) = o;
  }
}

extern "C" void kernel_launch(void* const* d_in, const int* in_sizes, int n_in,
                              void* d_out, int out_size, void* d_ws, size_t ws_size,
                              hipStream_t stream) {
  (void)n_in; (void)d_ws; (void)ws_size; (void)out_size;
  const float* in   = (const float*)d_in[0];
  const float* rate = (const float*)d_in[1];
  float*       out  = (float*)d_out;
  const int rows = in_sizes[0] / FEAT;    // 8192
  rsoftmax_kernel<<<dim3(rows), dim3(TPB), 0, stream>>>(in, rate, out);
}